// SpatialSoftmax_61649960567394
// MI455X (gfx1250) — compile-verified
//
#include <hip/hip_runtime.h>
#include <stdint.h>

#define BATCH   256
#define HDIM    512
#define WDIM    512
#define HW      (HDIM * WDIM)          // 262144
#define SPLITS  8
#define CHUNK   (HW / SPLITS)          // 32768 elements per block
#define THREADS 256
#define QUAD    8                      // floats per thread per tile
#define TILE    (THREADS * QUAD)       // 2048 floats = 8KB per tile
#define NT      (CHUNK / TILE)         // 16 tiles per block
#define NBUF    4                      // async ring depth (2 issues per tile -> 8 in flight)
#define TMIN_C  0.2f
#define TMAX_C  3.0f
#define NEG_BIG (-3.4e38f)
#define LOG2E   1.44269504088896340736f

// raw v_exp_f32 (base-2 exponential)
__device__ __forceinline__ float fexp2(float x) { return __builtin_amdgcn_exp2f(x); }

// Online-softmax merge of two partial states (base-2 domain).
__device__ __forceinline__ void osm_merge(float& m, float& s, float& sx, float& sy,
                                          float m2, float s2, float sx2, float sy2) {
    float mm = fmaxf(m, m2);
    float ra = fexp2(m  - mm);
    float rb = fexp2(m2 - mm);
    s  = s  * ra + s2  * rb;
    sx = sx * ra + sx2 * rb;
    sy = sy * ra + sy2 * rb;
    m  = mm;
}

__global__ __launch_bounds__(THREADS)
void spatial_softmax_pass1(const float* __restrict__ logits,
                           const float* __restrict__ temp_alpha,
                           float* __restrict__ partial) {
    __shared__ __align__(16) float tile[NBUF][TILE];
    __shared__ float red[THREADS / 32][4];

    const int blk   = blockIdx.x;            // 0 .. BATCH*SPLITS-1
    const int batch = blk / SPLITS;
    const int split = blk % SPLITS;
    const int tid   = threadIdx.x;

    // temperature = TMIN + (TMAX-TMIN)*sigmoid(alpha); fold log2(e) into scale
    const float a     = temp_alpha[0];
    const float sig   = 1.0f / (1.0f + __expf(-a));
    const float scale = LOG2E / (TMIN_C + (TMAX_C - TMIN_C) * sig);

    const uint64_t sbase     = (uint64_t)logits;
    const unsigned chunkByte = ((unsigned)batch * (unsigned)HW +
                                (unsigned)split * (unsigned)CHUNK) * 4u;
    const unsigned laneByte  = (unsigned)tid * (QUAD * 4u);   // 32B per lane per tile

    // ---- async pipeline prologue: fill NBUF tiles (2 x B128 per lane each) ----
    #pragma unroll
    for (int b = 0; b < NBUF; ++b) {
        unsigned ldsOff = (unsigned)(uint64_t)(&tile[b][tid * QUAD]);
        unsigned gOff   = chunkByte + (unsigned)b * (TILE * 4u) + laneByte;
        asm volatile("global_load_async_to_lds_b128 %0, %1, %4\n\t"
                     "global_load_async_to_lds_b128 %2, %3, %4"
                     :: "v"(ldsOff), "v"(gOff),
                        "v"(ldsOff + 16u), "v"(gOff + 16u),
                        "s"(sbase) : "memory");
    }

    float m = NEG_BIG, s = 0.f, sx = 0.f, sy = 0.f;

    // ---- streaming online-softmax reduction over NT tiles ---------------------
    for (int t = 0; t < NT; ++t) {
        // oldest tile's 2 async issues have landed (in-order completion):
        // steady-state outstanding = 2*NBUF, wait to <= 2*(NBUF-1)
        asm volatile("s_wait_asynccnt 6" ::: "memory");

        const float4 va = *(const float4*)(&tile[t % NBUF][tid * QUAD]);
        const float4 vb = *(const float4*)(&tile[t % NBUF][tid * QUAD + 4]);

        // refill the slot just consumed (dummy tail issues keep depth uniform)
        {
            unsigned ldsOff  = (unsigned)(uint64_t)(&tile[t % NBUF][tid * QUAD]);
            unsigned tileOff = (t + NBUF < NT) ? (unsigned)(t + NBUF) * (TILE * 4u) : 0u;
            unsigned gOff    = chunkByte + tileOff + laneByte;
            asm volatile("s_wait_dscnt 0\n\t"
                         "global_load_async_to_lds_b128 %0, %1, %4\n\t"
                         "global_load_async_to_lds_b128 %2, %3, %4"
                         :: "v"(ldsOff), "v"(gOff),
                            "v"(ldsOff + 16u), "v"(gOff + 16u),
                            "s"(sbase) : "memory");
        }

        const int k = split * CHUNK + t * TILE + tid * QUAD; // within-batch index
        float t0 = va.x * scale, t1 = va.y * scale, t2 = va.z * scale, t3 = va.w * scale;
        float t4 = vb.x * scale, t5 = vb.y * scale, t6 = vb.z * scale, t7 = vb.w * scale;

        float q = fmaxf(fmaxf(fmaxf(t0, t1), fmaxf(t2, t3)),
                        fmaxf(fmaxf(t4, t5), fmaxf(t6, t7)));
        if (q > m) {                        // rare after warm-up
            float r = fexp2(m - q);
            s *= r; sx *= r; sy *= r; m = q;
        }
        float e0 = fexp2(t0 - m), e1 = fexp2(t1 - m);
        float e2 = fexp2(t2 - m), e3 = fexp2(t3 - m);
        float e4 = fexp2(t4 - m), e5 = fexp2(t5 - m);
        float e6 = fexp2(t6 - m), e7 = fexp2(t7 - m);

        // octets never straddle a row (W % 8 == 0)
        float col0 = (float)(k & (WDIM - 1));
        float row  = (float)(k >> 9);       // k / WDIM
        float es   = ((e0 + e1) + (e2 + e3)) + ((e4 + e5) + (e6 + e7));
        s  += es;
        sy += es * row;
        float off = e1 + 2.f * e2 + 3.f * e3 + 4.f * e4 + 5.f * e5 + 6.f * e6 + 7.f * e7;
        sx += es * col0 + off;
    }

    // ---- wave32 reduction (online merge) --------------------------------------
    #pragma unroll
    for (int off = 16; off >= 1; off >>= 1) {
        float m2  = __shfl_xor(m,  off, 32);
        float s2  = __shfl_xor(s,  off, 32);
        float sx2 = __shfl_xor(sx, off, 32);
        float sy2 = __shfl_xor(sy, off, 32);
        osm_merge(m, s, sx, sy, m2, s2, sx2, sy2);
    }

    // ---- cross-wave reduction via LDS -----------------------------------------
    const int wave = tid >> 5, lane = tid & 31;
    if (lane == 0) {
        red[wave][0] = m; red[wave][1] = s; red[wave][2] = sx; red[wave][3] = sy;
    }
    __syncthreads();
    if (tid < (THREADS / 32)) {
        m = red[tid][0]; s = red[tid][1]; sx = red[tid][2]; sy = red[tid][3];
        #pragma unroll
        for (int off = 4; off >= 1; off >>= 1) {
            float m2  = __shfl_xor(m,  off, 8);
            float s2  = __shfl_xor(s,  off, 8);
            float sx2 = __shfl_xor(sx, off, 8);
            float sy2 = __shfl_xor(sy, off, 8);
            osm_merge(m, s, sx, sy, m2, s2, sx2, sy2);
        }
        if (tid == 0) {
            float* p = partial + (size_t)blk * 4;
            p[0] = m; p[1] = s; p[2] = sx; p[3] = sy;
        }
    }
}

__global__ __launch_bounds__(32)
void spatial_softmax_pass2(const float* __restrict__ partial,
                           float* __restrict__ out) {
    const int b    = blockIdx.x;     // batch
    const int lane = threadIdx.x;    // lanes 0..7 carry data

    float m = NEG_BIG, s = 0.f, sx = 0.f, sy = 0.f;
    if (lane < SPLITS) {
        const float* p = partial + ((size_t)b * SPLITS + lane) * 4;
        m = p[0]; s = p[1]; sx = p[2]; sy = p[3];
    }
    #pragma unroll
    for (int off = 4; off >= 1; off >>= 1) {
        float m2  = __shfl_xor(m,  off, 8);
        float s2  = __shfl_xor(s,  off, 8);
        float sx2 = __shfl_xor(sx, off, 8);
        float sy2 = __shfl_xor(sy, off, 8);
        osm_merge(m, s, sx, sy, m2, s2, sx2, sy2);
    }
    if (lane == 0) {
        // coords accumulated as integer col/row; linspace scale = 1/(dim-1)
        float inv = 1.0f / s;
        out[b * 2 + 0] = sx * inv * (1.0f / (float)(WDIM - 1));
        out[b * 2 + 1] = sy * inv * (1.0f / (float)(HDIM - 1));
    }
}

extern "C" void kernel_launch(void* const* d_in, const int* in_sizes, int n_in,
                              void* d_out, int out_size, void* d_ws, size_t ws_size,
                              hipStream_t stream) {
    const float* logits     = (const float*)d_in[0];
    const float* temp_alpha = (const float*)d_in[1];
    float*       out        = (float*)d_out;
    float*       partial    = (float*)d_ws;   // BATCH*SPLITS*4 floats = 32 KB

    spatial_softmax_pass1<<<BATCH * SPLITS, THREADS, 0, stream>>>(logits, temp_alpha, partial);
    spatial_softmax_pass2<<<BATCH, 32, 0, stream>>>(partial, out);
}